// GaussianKernel_29334626632409
// MI455X (gfx1250) — compile-verified
//
#include <hip/hip_runtime.h>

// Gaussian RBF kernel matrix: out[n,m] = exp(-0.5 * max(|s_n|^2 + |t_m|^2 - 2 s_n.t_m, 0))
// N = M = 8192, K = 64, f32. Store-bound (256 MB out); GEMM via V_WMMA_F32_16X16X4_F32.

typedef __attribute__((ext_vector_type(2))) float v2f;
typedef __attribute__((ext_vector_type(8))) float v8f;

#define NROWS 8192
#define KDIM  64

// ---------------------------------------------------------------------------
// Kernel 1: row squared-norms for src and tar -> workspace
// ---------------------------------------------------------------------------
__global__ __launch_bounds__(256) void gk_norms(const float* __restrict__ src,
                                                const float* __restrict__ tar,
                                                float* __restrict__ srcn,
                                                float* __restrict__ tarn) {
    int i = blockIdx.x * blockDim.x + threadIdx.x;
    if (i >= 2 * NROWS) return;
    const float* base = (i < NROWS) ? src : tar;
    float* outp       = (i < NROWS) ? srcn : tarn;
    int r             = (i < NROWS) ? i : i - NROWS;
    const float4* p = (const float4*)(base + (size_t)r * KDIM);
    float s = 0.0f;
#pragma unroll
    for (int j = 0; j < KDIM / 4; ++j) {
        float4 v = p[j];
        s += v.x * v.x + v.y * v.y + v.z * v.z + v.w * v.w;
    }
    outp[r] = s;
}

// ---------------------------------------------------------------------------
// Kernel 2: WMMA f32 GEMM strip + fused norm/exp epilogue.
// Block = 256 threads = 8 waves. Each wave -> 16 rows x 64 cols of output
// (4 accumulators). Block -> 128 rows x 64 cols. Grid = (8192/64, 8192/128).
// ---------------------------------------------------------------------------
__global__ __launch_bounds__(256) void gk_wmma(const float* __restrict__ src,
                                               const float* __restrict__ tar,
                                               const float* __restrict__ srcn,
                                               const float* __restrict__ tarn,
                                               float* __restrict__ out) {
    const int lane = threadIdx.x & 31;
    const int wave = threadIdx.x >> 5;
    const int lid  = lane & 15;     // row/col within tile
    const int hi   = lane >> 4;     // 0: K={0,1}, 1: K={2,3} within a K-step

    const int rowBase = blockIdx.y * 128 + wave * 16;  // src rows for this wave
    const int colBase = blockIdx.x * 64;               // tar cols for this block

    // A operand: lane holds src[rowBase+lid][k0 + 2*hi + {0,1}]
    const float* aptr = src + (size_t)(rowBase + lid) * KDIM + 2 * hi;
    // B operand (tile t): lane holds tar[colBase + 16*t + lid][k0 + 2*hi + {0,1}]
    const float* bptr = tar + (size_t)(colBase + lid) * KDIM + 2 * hi;

    v8f acc0 = {}, acc1 = {}, acc2 = {}, acc3 = {};

#pragma unroll
    for (int k0 = 0; k0 < KDIM; k0 += 4) {
        v2f a  = *(const v2f*)(aptr + k0);
        v2f b0 = *(const v2f*)(bptr + k0);
        v2f b1 = *(const v2f*)(bptr + 16 * KDIM + k0);
        v2f b2 = *(const v2f*)(bptr + 32 * KDIM + k0);
        v2f b3 = *(const v2f*)(bptr + 48 * KDIM + k0);
        // 8 args: (neg_a, A, neg_b, B, c_mod, C, reuse_a, reuse_b)
        acc0 = __builtin_amdgcn_wmma_f32_16x16x4_f32(false, a, false, b0,
                                                     (short)0, acc0, false, false);
        acc1 = __builtin_amdgcn_wmma_f32_16x16x4_f32(false, a, false, b1,
                                                     (short)0, acc1, false, false);
        acc2 = __builtin_amdgcn_wmma_f32_16x16x4_f32(false, a, false, b2,
                                                     (short)0, acc2, false, false);
        acc3 = __builtin_amdgcn_wmma_f32_16x16x4_f32(false, a, false, b3,
                                                     (short)0, acc3, false, false);
    }

    // Epilogue. D layout: VGPR r, lanes 0-15 -> (M=r, N=lane), lanes 16-31 -> (M=r+8, N=lane-16).
    const int mrow = rowBase + 8 * hi;   // first output row this half-wave owns

    float sq[8];
#pragma unroll
    for (int r = 0; r < 8; ++r) sq[r] = srcn[mrow + r];   // broadcast within half-wave (cached)

    float tq0 = tarn[colBase + 0  + lid];
    float tq1 = tarn[colBase + 16 + lid];
    float tq2 = tarn[colBase + 32 + lid];
    float tq3 = tarn[colBase + 48 + lid];

#pragma unroll
    for (int r = 0; r < 8; ++r) {
        float* orow = out + (size_t)(mrow + r) * NROWS + colBase + lid;
        float d0 = fmaxf(sq[r] + tq0 - 2.0f * acc0[r], 0.0f);
        float d1 = fmaxf(sq[r] + tq1 - 2.0f * acc1[r], 0.0f);
        float d2 = fmaxf(sq[r] + tq2 - 2.0f * acc2[r], 0.0f);
        float d3 = fmaxf(sq[r] + tq3 - 2.0f * acc3[r], 0.0f);
        // 256 MB output stream > 192 MB L2: bypass with non-temporal stores.
        __builtin_nontemporal_store(__expf(-0.5f * d0), orow + 0);
        __builtin_nontemporal_store(__expf(-0.5f * d1), orow + 16);
        __builtin_nontemporal_store(__expf(-0.5f * d2), orow + 32);
        __builtin_nontemporal_store(__expf(-0.5f * d3), orow + 48);
    }
}

// ---------------------------------------------------------------------------
extern "C" void kernel_launch(void* const* d_in, const int* in_sizes, int n_in,
                              void* d_out, int out_size, void* d_ws, size_t ws_size,
                              hipStream_t stream) {
    const float* src = (const float*)d_in[0];
    const float* tar = (const float*)d_in[1];
    float* out  = (float*)d_out;
    float* srcn = (float*)d_ws;            // 8192 floats
    float* tarn = srcn + NROWS;            // 8192 floats (total 64 KB of d_ws)

    gk_norms<<<(2 * NROWS + 255) / 256, 256, 0, stream>>>(src, tar, srcn, tarn);

    dim3 grid(NROWS / 64, NROWS / 128);    // (128, 64)
    gk_wmma<<<grid, 256, 0, stream>>>(src, tar, srcn, tarn, out);
}